// SaliencyEvaluator_PSR_36567351558144
// MI455X (gfx1250) — compile-verified
//
#include <hip/hip_runtime.h>
#include <hip/hip_bf16.h>

// Problem dims (match reference setup_inputs): B=32, C=256, H=W=96
#define BATCH 32
#define CHANS 256
#define HDIM 96
#define WDIM 96
#define HW (HDIM * WDIM)          // 9216 floats per row
#define THREADS 256
#define F4_PER_THREAD (HW / 4 / THREADS)  // 2304/256 = 9

typedef __attribute__((ext_vector_type(2))) float v2f;
typedef __attribute__((ext_vector_type(8))) float v8f;

// Sum the 64 per-lane partials of a wave (2 per lane, packed as the 16x4 f32
// A-matrix) with one V_WMMA_F32_16X16X4_F32 against an all-ones B.
// D[i][j] = rowsum_i for every column j; a lane's 8 C regs hold D[r..][lane-col],
// so laneSum + shfl_xor(laneSum,16) == total of all 64 partials, on every lane.
__device__ __forceinline__ float wave_sum_wmma(float p0, float p1) {
    v2f a;    a[0] = p0;   a[1] = p1;
    v2f ones; ones[0] = 1.0f; ones[1] = 1.0f;
    v8f c = {};
    c = __builtin_amdgcn_wmma_f32_16x16x4_f32(
            /*neg_a=*/false, a, /*neg_b=*/false, ones,
            /*c_mod=*/(short)0, c, /*reuse_a=*/false, /*reuse_b=*/false);
    float s = ((c[0] + c[1]) + (c[2] + c[3])) + ((c[4] + c[5]) + (c[6] + c[7]));
    s += __shfl_xor(s, 16, 32);
    return s;   // full wave total (valid in every lane)
}

__global__ __launch_bounds__(THREADS)
void psr_rows_kernel(const float* __restrict__ cv,
                     const int*   __restrict__ peaks,
                     float*       __restrict__ psr) {
    const int row = blockIdx.x;          // b*C + c
    const int tid = threadIdx.x;
    const float* base = cv + (size_t)row * HW;
    const float4* p4 = (const float4*)base;

    // ---- streaming pass: per-thread sum / sumsq / max over 36 elements ----
    float s0 = 0.f, s1 = 0.f;            // partial sums (2 accumulators)
    float q0 = 0.f, q1 = 0.f;            // partial sums of squares
    float pm = -3.402823466e38f;
    #pragma unroll
    for (int i = 0; i < F4_PER_THREAD; ++i) {
        float4 v = p4[i * THREADS + tid];          // coalesced b128 loads
        s0 += v.x; s1 += v.y; s0 += v.z; s1 += v.w;
        q0 = fmaf(v.x, v.x, q0); q1 = fmaf(v.y, v.y, q1);
        q0 = fmaf(v.z, v.z, q0); q1 = fmaf(v.w, v.w, q1);
        pm = fmaxf(pm, fmaxf(fmaxf(v.x, v.y), fmaxf(v.z, v.w)));
    }

    // ---- wave-level reduction: WMMA for sums, shfl tree for max ----
    // (all 256 threads active here: EXEC all ones as WMMA requires)
    float wsum = wave_sum_wmma(s0, s1);
    float wsq  = wave_sum_wmma(q0, q1);
    #pragma unroll
    for (int off = 16; off > 0; off >>= 1)
        pm = fmaxf(pm, __shfl_xor(pm, off, 32));

    __shared__ float ls[8], lq[8], lm[8];
    const int wave = tid >> 5, lane = tid & 31;
    if (lane == 0) { ls[wave] = wsum; lq[wave] = wsq; lm[wave] = pm; }
    __syncthreads();

    // ---- window (mainlobe) contribution: tid<25 all live in wave 0 ----
    if (tid < 32) {
        float winS = 0.f, winQ = 0.f;
        int   winN = 0;
        if (tid < 25) {
            const int dy = tid / 5 - 2, dx = tid % 5 - 2;
            const int py = peaks[row * 2 + 0];
            const int px = peaks[row * 2 + 1];
            const int y = py + dy, x = px + dx;
            // clipped duplicates are exactly the out-of-bounds offsets
            if (y >= 0 && y < HDIM && x >= 0 && x < WDIM) {
                const float v = base[y * WDIM + x];
                winS = v; winQ = v * v; winN = 1;
            }
        }
        #pragma unroll
        for (int off = 16; off > 0; off >>= 1) {
            winS += __shfl_xor(winS, off, 32);
            winQ += __shfl_xor(winQ, off, 32);
            winN += __shfl_xor(winN, off, 32);
        }
        if (tid == 0) {
            float ts = 0.f, tq = 0.f, tm = -3.402823466e38f;
            #pragma unroll
            for (int w = 0; w < 8; ++w) {
                ts += ls[w]; tq += lq[w]; tm = fmaxf(tm, lm[w]);
            }
            const float n        = (float)(HW - winN);
            const float sum_side = ts - winS;
            const float sq_side  = tq - winQ;
            const float mean     = sum_side / n;
            // sum_side((cv-mean)^2) = sq_side - n*mean^2
            const float var      = (sq_side - n * mean * mean) / (n - 1.0f);
            psr[row] = (tm - mean) / var;
        }
    }
}

__global__ __launch_bounds__(THREADS)
void saliency_norm_kernel(const float* __restrict__ psr,
                          float*       __restrict__ out) {
    const int b = blockIdx.x, t = threadIdx.x;
    const float p = psr[b * CHANS + t];

    float s = p;
    #pragma unroll
    for (int off = 16; off > 0; off >>= 1)
        s += __shfl_xor(s, off, 32);

    __shared__ float ws[8];
    if ((t & 31) == 0) ws[t >> 5] = s;
    __syncthreads();

    float tot = 0.f;
    #pragma unroll
    for (int w = 0; w < 8; ++w) tot += ws[w];
    const float mean = tot * (1.0f / (float)CHANS);
    out[b * CHANS + t] = p / (mean + 1e-8f);
}

extern "C" void kernel_launch(void* const* d_in, const int* in_sizes, int n_in,
                              void* d_out, int out_size, void* d_ws, size_t ws_size,
                              hipStream_t stream) {
    const float* cv    = (const float*)d_in[0];   // [B,C,H,W] fp32
    const int*   peaks = (const int*)  d_in[1];   // [B,C,2] int32 (y,x)
    float*       out   = (float*)d_out;           // [B,C] fp32
    float*       psr   = (float*)d_ws;            // scratch: B*C floats

    psr_rows_kernel<<<BATCH * CHANS, THREADS, 0, stream>>>(cv, peaks, psr);
    saliency_norm_kernel<<<BATCH, THREADS, 0, stream>>>(psr, out);
}